// CovMeAggLayer_52518860095501
// MI455X (gfx1250) — compile-verified
//
#include <hip/hip_runtime.h>

#define NN 40000
#define NE 640000
#define DD 128
#define BN_EPS 1e-5f
#define SLOPE 0.01f

typedef __attribute__((ext_vector_type(2))) float v2f;
typedef __attribute__((ext_vector_type(4))) float v4f;
typedef __attribute__((ext_vector_type(8))) float v8f;

// ---------------------------------------------------------------------------
// Kernel 1: accum = node_feat (residual pre-added), zero BN stat accumulators
// ---------------------------------------------------------------------------
__global__ void init_kernel(const float* __restrict__ node_feat,
                            float* __restrict__ accum,
                            float* __restrict__ stats) {
    int i = blockIdx.x * blockDim.x + threadIdx.x;
    const int total4 = NN * DD / 4;
    if (i < total4) {
        reinterpret_cast<v4f*>(accum)[i] =
            reinterpret_cast<const v4f*>(node_feat)[i];
    }
    if (blockIdx.x == 0 && threadIdx.x < 2 * DD) stats[threadIdx.x] = 0.0f;
}

// ---------------------------------------------------------------------------
// Kernel 2: per-edge message + scatter-add. One wave per edge, float4/lane.
// edge_feat (328 MB one-shot stream) is loaded NON-TEMPORAL so it does not
// evict the L2-resident node_feat gather set / atomic accumulator (~41 MB
// combined vs 192 MB L2). Gathers and the 82M f32 atomics stay L2-side.
// ---------------------------------------------------------------------------
__global__ void edge_kernel(const float* __restrict__ node_feat,
                            const float* __restrict__ edge_feat,
                            const int*   __restrict__ esrc,
                            const int*   __restrict__ edst,
                            float*       __restrict__ accum) {
    int gid  = blockIdx.x * blockDim.x + threadIdx.x;
    int e    = gid >> 5;            // one wave32 per edge
    if (e >= NE) return;
    int lane = threadIdx.x & 31;    // lane handles dims [4*lane, 4*lane+4)
    int s = __builtin_nontemporal_load(esrc + e);
    int d = __builtin_nontemporal_load(edst + e);

    v4f nf = *reinterpret_cast<const v4f*>(node_feat + (size_t)s * DD + lane * 4);
    v4f ef = __builtin_nontemporal_load(
        reinterpret_cast<const v4f*>(edge_feat + (size_t)e * DD + lane * 4));

    v4f m;
    m.x = fmaxf(nf.x + ef.x, 0.0f);
    m.y = fmaxf(nf.y + ef.y, 0.0f);
    m.z = fmaxf(nf.z + ef.z, 0.0f);
    m.w = fmaxf(nf.w + ef.w, 0.0f);

    float* out = accum + (size_t)d * DD + lane * 4;
    atomicAdd(out + 0, m.x);
    atomicAdd(out + 1, m.y);
    atomicAdd(out + 2, m.z);
    atomicAdd(out + 3, m.w);
}

// ---------------------------------------------------------------------------
// Kernel 3: y = leakyrelu(rst @ W^T + b) via V_WMMA_F32_16X16X4_F32.
// Block = 256 threads = 8 waves; wave w owns output tile [16 rows x 16 cols],
// covering all 8 n-tiles of the 128-wide output. K=128 -> 32 chained WMMAs.
//
// Fragment layouts (ISA 7.12.2, 32-bit operands):
//   A 16x4 : lane(l) holds A[l&15][k0 + 2*(l>>4) + {0,1}]      (float2)
//   B 4x16 : lane(l) holds B[k0 + 2*(l>>4) + {0,1}][l&15]      (float2)
//   C 16x16: vgpr v, lanes 0-15 -> M=v, N=l; lanes 16-31 -> M=v+8, N=l-16
// B = W^T  =>  B[k][n] = W[n][k]  => contiguous float2 from row-major W.
// ---------------------------------------------------------------------------
__global__ void gemm_kernel(const float* __restrict__ rst,
                            const float* __restrict__ Wm,
                            const float* __restrict__ bias,
                            float*       __restrict__ y) {
    int wave = threadIdx.x >> 5;     // n-tile index 0..7
    int lane = threadIdx.x & 31;
    int half = lane >> 4;            // 0 or 1
    int l16  = lane & 15;

    int row = blockIdx.x * 16 + l16; // A fragment row (M = l16)
    int col = wave * 16 + l16;       // B fragment col (N = l16)

    const float* arow = rst + (size_t)row * DD;
    const float* brow = Wm  + (size_t)col * DD;

    v8f acc = {0.f, 0.f, 0.f, 0.f, 0.f, 0.f, 0.f, 0.f};

#pragma unroll
    for (int k0 = 0; k0 < DD; k0 += 4) {
        int k  = k0 + half * 2;
        v2f a  = *reinterpret_cast<const v2f*>(arow + k);
        v2f bb = *reinterpret_cast<const v2f*>(brow + k);
        acc = __builtin_amdgcn_wmma_f32_16x16x4_f32(
            /*neg_a=*/false, a, /*neg_b=*/false, bb,
            /*c_mod=*/(short)0, acc, /*reuse_a=*/false, /*reuse_b=*/false);
    }

    float bv = bias[col];
#pragma unroll
    for (int v = 0; v < 8; ++v) {
        int r = blockIdx.x * 16 + v + 8 * half;  // C/D layout row
        float val = acc[v] + bv;
        val = (val >= 0.0f) ? val : SLOPE * val; // LeakyReLU fused
        y[(size_t)r * DD + col] = val;
    }
}

// ---------------------------------------------------------------------------
// Kernel 4: per-column sum / sumsq (coalesced row-major reads, L2-resident y)
// ---------------------------------------------------------------------------
__global__ void stats_kernel(const float* __restrict__ y,
                             float* __restrict__ stats) {
    int col = threadIdx.x;           // blockDim.x == 128
    float s = 0.0f, ss = 0.0f;
    for (int r = blockIdx.x; r < NN; r += gridDim.x) {
        float v = y[(size_t)r * DD + col];
        s  += v;
        ss += v * v;
    }
    atomicAdd(&stats[col], s);
    atomicAdd(&stats[DD + col], ss);
}

// ---------------------------------------------------------------------------
// Kernel 5: fold BN into per-column scale/shift
// ---------------------------------------------------------------------------
__global__ void finalize_kernel(const float* __restrict__ stats,
                                const float* __restrict__ gamma,
                                const float* __restrict__ beta,
                                float* __restrict__ ss) {
    int c = threadIdx.x;             // 128 threads
    const float inv_n = 1.0f / (float)NN;
    float mean = stats[c] * inv_n;
    float var  = stats[DD + c] * inv_n - mean * mean;
    var = fmaxf(var, 0.0f);
    float sc = gamma[c] * rsqrtf(var + BN_EPS);
    ss[c]      = sc;
    ss[DD + c] = beta[c] - mean * sc;
}

// ---------------------------------------------------------------------------
// Kernel 6: out = y * scale[col] + shift[col]. In-place on d_out; final
// stores are non-temporal (nothing re-reads d_out on-device afterwards).
// ---------------------------------------------------------------------------
__global__ void norm_kernel(float* __restrict__ y,
                            const float* __restrict__ ss) {
    int i = blockIdx.x * blockDim.x + threadIdx.x;   // float4 index
    const int total4 = NN * DD / 4;
    if (i >= total4) return;
    int c = (i * 4) & (DD - 1);
    v4f v = reinterpret_cast<const v4f*>(y)[i];
    v.x = v.x * ss[c + 0] + ss[DD + c + 0];
    v.y = v.y * ss[c + 1] + ss[DD + c + 1];
    v.z = v.z * ss[c + 2] + ss[DD + c + 2];
    v.w = v.w * ss[c + 3] + ss[DD + c + 3];
    __builtin_nontemporal_store(v, reinterpret_cast<v4f*>(y) + i);
}

extern "C" void kernel_launch(void* const* d_in, const int* in_sizes, int n_in,
                              void* d_out, int out_size, void* d_ws, size_t ws_size,
                              hipStream_t stream) {
    const float* node_feat = (const float*)d_in[0];
    const float* edge_feat = (const float*)d_in[1];
    const int*   esrc      = (const int*)d_in[2];
    const int*   edst      = (const int*)d_in[3];
    const float* Wm        = (const float*)d_in[4];
    const float* bias      = (const float*)d_in[5];
    const float* gamma     = (const float*)d_in[6];
    const float* beta      = (const float*)d_in[7];
    float* out = (float*)d_out;      // doubles as y buffer (fully rewritten)

    char*  ws    = (char*)d_ws;
    float* accum = (float*)ws;                                     // NN*DD floats
    float* stats = (float*)(ws + (size_t)NN * DD * sizeof(float)); // 2*DD
    float* ss    = stats + 2 * DD;                                 // 2*DD

    const int elem4 = NN * DD / 4;                                 // 1,280,000

    init_kernel<<<(elem4 + 255) / 256, 256, 0, stream>>>(node_feat, accum, stats);
    edge_kernel<<<(NE * 32) / 256, 256, 0, stream>>>(node_feat, edge_feat,
                                                     esrc, edst, accum);
    gemm_kernel<<<NN / 16, 256, 0, stream>>>(accum, Wm, bias, out);
    stats_kernel<<<500, DD, 0, stream>>>(out, stats);
    finalize_kernel<<<1, DD, 0, stream>>>(stats, gamma, beta, ss);
    norm_kernel<<<(elem4 + 255) / 256, 256, 0, stream>>>(out, ss);
}